// HeteroSAGEConv_16767552323880
// MI455X (gfx1250) — compile-verified
//
#include <hip/hip_runtime.h>
#include <hip/hip_bf16.h>

// ---------------------------------------------------------------------------
// HeteroSAGEConv for MI455X (gfx1250), wave32 + WMMA f32 16x16x4.
//   - Weights pre-transposed in global ws so B fragments are K-contiguous
//     (single ds_load_b64, conflict-free with 132-dword LDS row pitch).
//   - W staged to LDS with global_load_async_to_lds_b128 (ASYNCcnt path).
//   - Scatter-mean via fp32 atomics; accumulators (~102 MB) are L2-resident.
// ---------------------------------------------------------------------------

#define N_USER  100000
#define N_SPOT  50000
#define D       128
#define E_VISIT 500000
#define E_REV   500000
#define E_NEAR  200000

#define LDS_PITCH 132                      // dwords; 132 % 64 banks = 4 -> no conflicts
#define LDS_BYTES (D * LDS_PITCH * 4)      // 67,584 B dynamic LDS

typedef __attribute__((ext_vector_type(2))) float v2f;
typedef __attribute__((ext_vector_type(8))) float v8f;

// ---------------------------------------------------------------------------
// Out[c*128 + r] = A[r*128 + c] (+ B[r*128 + c] if B != null)
// ---------------------------------------------------------------------------
__global__ void transpose_add_kernel(const float* __restrict__ a,
                                     const float* __restrict__ b,
                                     float* __restrict__ o) {
    int i = blockIdx.x * blockDim.x + threadIdx.x;
    if (i < D * D) {
        int r = i >> 7, c = i & 127;
        float v = a[i];
        if (b) v += b[i];
        o[c * D + r] = v;
    }
}

// ---------------------------------------------------------------------------
// Y[nRows x 128] = X[nRows x 128] @ W,  W given TRANSPOSED (Wt[n][k] = W[k][n])
// Optional fused epilogue:
//   if (sum1) y = relu(scale * (y + sum1[r]/max(cnt1[r],1) [+ sum2[r]/max(cnt2[r],1)]))
//
// Block: 256 threads = 8 waves; each wave owns a 16-row x 128-col strip.
// Wt staged into padded LDS via async global->LDS (b128 per lane).
// ---------------------------------------------------------------------------
__global__ __launch_bounds__(256) void
gemm128_wmma_kernel(const float* __restrict__ X,
                    const float* __restrict__ Wt,
                    float* __restrict__ Y,
                    int nRows,
                    const float* __restrict__ sum1,
                    const float* __restrict__ cnt1,
                    const float* __restrict__ sum2,
                    const float* __restrict__ cnt2,
                    float scale) {
    extern __shared__ float Wl[];          // [128][LDS_PITCH]

    const int tid = threadIdx.x;

    // --- Async stage: 4096 x 16B chunks, 16 per thread, row-padded in LDS ---
    {
        const unsigned lbase = (unsigned)(size_t)&Wl[0];   // low 32 bits = LDS offset
        #pragma unroll
        for (int i = 0; i < 16; ++i) {
            const int chunk = i * 256 + tid;               // 0..4095
            const int r  = chunk >> 5;                     // row (32 chunks/row)
            const int c4 = chunk & 31;                     // 16B chunk within row
            unsigned long long gaddr =
                (unsigned long long)(size_t)Wt + (unsigned)((r * D + c4 * 4) * 4);
            unsigned laddr = lbase + (unsigned)((r * LDS_PITCH + c4 * 4) * 4);
            asm volatile("global_load_async_to_lds_b128 %0, %1, off"
                         :: "v"(laddr), "v"(gaddr) : "memory");
        }
        asm volatile("s_wait_asynccnt 0x0" ::: "memory");
    }
    __syncthreads();

    const int lane    = tid & 31;
    const int wave    = tid >> 5;          // 0..7
    const int lh      = lane >> 4;         // half-wave select
    const int ln      = lane & 15;
    const int rowBase = blockIdx.x * 128 + wave * 16;

    if (rowBase >= nRows) return;          // whole-wave exit (after barrier)

    // A source row, clamped so all 32 lanes stay active (EXEC must be all 1s)
    int arow = rowBase + ln;
    if (arow > nRows - 1) arow = nRows - 1;
    const float* xrow = X + (size_t)arow * D;

    v8f acc[8];
    #pragma unroll
    for (int nb = 0; nb < 8; ++nb) acc[nb] = (v8f)(0.0f);

    // K loop: 32 WMMA steps of K=4. A frag: lane ln holds row, VGPR0/1 = K=ka,ka+1.
    // B frag from transposed LDS: contiguous b64 at Wl[n*PITCH + ka].
    for (int k0 = 0; k0 < D; k0 += 4) {
        const int ka = k0 + lh * 2;        // even -> 8B aligned LDS loads
        v2f a;
        a.x = xrow[ka];
        a.y = xrow[ka + 1];
        const float* wb = &Wl[(size_t)ln * LDS_PITCH + ka];
        #pragma unroll
        for (int nb = 0; nb < 8; ++nb) {
            v2f b = *(const v2f*)(wb + nb * 16 * LDS_PITCH);
            acc[nb] = __builtin_amdgcn_wmma_f32_16x16x4_f32(
                false, a, false, b, (short)0, acc[nb], false, false);
        }
    }

    // C/D layout: VGPR i -> row rowBase + lh*8 + i, col nb*16 + ln.
    #pragma unroll
    for (int i = 0; i < 8; ++i) {
        const int r = rowBase + lh * 8 + i;
        if (r < nRows) {
            float inv1 = 0.0f, inv2 = 0.0f;
            if (sum1) {
                inv1 = 1.0f / fmaxf(cnt1[r], 1.0f);
                if (sum2) inv2 = 1.0f / fmaxf(cnt2[r], 1.0f);
            }
            #pragma unroll
            for (int nb = 0; nb < 8; ++nb) {
                const int c = nb * 16 + ln;
                float v = acc[nb][i];
                if (sum1) {
                    v += sum1[(size_t)r * D + c] * inv1;
                    if (sum2) v += sum2[(size_t)r * D + c] * inv2;
                    v *= scale;
                    v = fmaxf(v, 0.0f);
                }
                Y[(size_t)r * D + c] = v;
            }
        }
    }
}

// ---------------------------------------------------------------------------
// Edge scatter: v = dropout(P[src[e]][t]); atomicAdd into sum[dst[e]][t];
// one lane per edge bumps cnt[dst[e]]. 2 edges / 256-thread block.
// ---------------------------------------------------------------------------
__global__ __launch_bounds__(256) void
scatter_mean_acc_kernel(const float* __restrict__ P,
                        const int* __restrict__ src,
                        const int* __restrict__ dst,
                        float* __restrict__ sum,
                        float* __restrict__ cnt,
                        int nEdges, unsigned seed) {
    const int e = blockIdx.x * 2 + (threadIdx.x >> 7);
    const int t = threadIdx.x & 127;
    if (e >= nEdges) return;

    const int s = src[e];
    const int d = dst[e];
    float v = P[(size_t)s * D + t];

    // Deterministic per-(edge, element) hash -> uniform [0,1)
    unsigned h = (unsigned)e * 2654435761u ^ ((unsigned)t * 0x9E3779B9u) ^ seed;
    h ^= h >> 16; h *= 0x85ebca6bu;
    h ^= h >> 13; h *= 0xc2b2ae35u;
    h ^= h >> 16;
    const float u = (float)(h >> 8) * (1.0f / 16777216.0f);
    v = (u < 0.9f) ? v * (1.0f / 0.9f) : 0.0f;   // keep w.p. 0.9, scale 1/0.9

    atomicAdd(&sum[(size_t)d * D + t], v);
    if (t == 0) atomicAdd(&cnt[d], 1.0f);        // counts all edges (pre-dropout)
}

// ---------------------------------------------------------------------------
// Launch
// ---------------------------------------------------------------------------
extern "C" void kernel_launch(void* const* d_in, const int* in_sizes, int n_in,
                              void* d_out, int out_size, void* d_ws, size_t ws_size,
                              hipStream_t stream) {
    const float* x_user      = (const float*)d_in[0];
    const float* x_spot      = (const float*)d_in[1];
    const float* W_visit_src = (const float*)d_in[2];
    const float* W_visit_tgt = (const float*)d_in[3];
    const float* W_rev_src   = (const float*)d_in[4];
    const float* W_rev_tgt   = (const float*)d_in[5];
    const float* W_near_src  = (const float*)d_in[6];
    const float* W_near_tgt  = (const float*)d_in[7];
    const int*   visit_src   = (const int*)d_in[8];
    const int*   visit_dst   = (const int*)d_in[9];
    const int*   rev_src     = (const int*)d_in[10];
    const int*   rev_dst     = (const int*)d_in[11];
    const int*   near_src    = (const int*)d_in[12];
    const int*   near_dst    = (const int*)d_in[13];

    float* out_user = (float*)d_out;                     // [N_USER x D]
    float* out_spot = out_user + (size_t)N_USER * D;     // [N_SPOT x D]

    // Workspace partition
    const size_t PU = (size_t)N_USER * D;
    const size_t PS = (size_t)N_SPOT * D;
    float* P_vu       = (float*)d_ws;       // x_user @ W_visit_src
    float* P_sr       = P_vu + PU;          // x_spot @ W_rev_src
    float* P_sn       = P_sr + PS;          // x_spot @ W_near_src
    float* sum_user   = P_sn + PS;          // zeroed scatter sums
    float* sum_visit  = sum_user + PU;
    float* sum_near   = sum_visit + PS;
    float* cnt_user   = sum_near + PS;
    float* cnt_visit  = cnt_user + N_USER;
    float* cnt_near   = cnt_visit + N_SPOT;
    float* Wt_vsrc    = cnt_near + N_SPOT;  // transposed weights (K-contiguous rows)
    float* Wt_rsrc    = Wt_vsrc + D * D;
    float* Wt_nsrc    = Wt_rsrc + D * D;
    float* Wt_rtgt    = Wt_nsrc + D * D;
    float* Wt_comb    = Wt_rtgt + D * D;    // (W_visit_tgt + W_near_tgt)^T

    // Zero sums + counts (contiguous), capture-safe async memset.
    const size_t zeroFloats = PU + PS + PS + N_USER + N_SPOT + N_SPOT;
    hipMemsetAsync(sum_user, 0, zeroFloats * sizeof(float), stream);

    // Pre-transpose weights (and fold the two spot-target matrices).
    const int gW = (D * D + 255) / 256;
    transpose_add_kernel<<<gW, 256, 0, stream>>>(W_visit_src, nullptr, Wt_vsrc);
    transpose_add_kernel<<<gW, 256, 0, stream>>>(W_rev_src,   nullptr, Wt_rsrc);
    transpose_add_kernel<<<gW, 256, 0, stream>>>(W_near_src,  nullptr, Wt_nsrc);
    transpose_add_kernel<<<gW, 256, 0, stream>>>(W_rev_tgt,   nullptr, Wt_rtgt);
    transpose_add_kernel<<<gW, 256, 0, stream>>>(W_visit_tgt, W_near_tgt, Wt_comb);

    // Source projections.
    const int gUser = (N_USER + 127) / 128;
    const int gSpot = (N_SPOT + 127) / 128;
    gemm128_wmma_kernel<<<gUser, 256, LDS_BYTES, stream>>>(
        x_user, Wt_vsrc, P_vu, N_USER, nullptr, nullptr, nullptr, nullptr, 1.0f);
    gemm128_wmma_kernel<<<gSpot, 256, LDS_BYTES, stream>>>(
        x_spot, Wt_rsrc, P_sr, N_SPOT, nullptr, nullptr, nullptr, nullptr, 1.0f);
    gemm128_wmma_kernel<<<gSpot, 256, LDS_BYTES, stream>>>(
        x_spot, Wt_nsrc, P_sn, N_SPOT, nullptr, nullptr, nullptr, nullptr, 1.0f);

    // Edge scatters (dropout + sum + count).
    scatter_mean_acc_kernel<<<(E_VISIT + 1) / 2, 256, 0, stream>>>(
        P_vu, visit_src, visit_dst, sum_visit, cnt_visit, E_VISIT, 0x1234567u);
    scatter_mean_acc_kernel<<<(E_REV + 1) / 2, 256, 0, stream>>>(
        P_sr, rev_src, rev_dst, sum_user, cnt_user, E_REV, 0x89ABCDEu);
    scatter_mean_acc_kernel<<<(E_NEAR + 1) / 2, 256, 0, stream>>>(
        P_sn, near_src, near_dst, sum_near, cnt_near, E_NEAR, 0xF00DFACu);

    // Finalize: target projection + mean(s) + scale + ReLU fused in epilogue.
    gemm128_wmma_kernel<<<gUser, 256, LDS_BYTES, stream>>>(
        x_user, Wt_rtgt, out_user, N_USER,
        sum_user, cnt_user, nullptr, nullptr, 1.0f);
    gemm128_wmma_kernel<<<gSpot, 256, LDS_BYTES, stream>>>(
        x_spot, Wt_comb, out_spot, N_SPOT,
        sum_visit, cnt_visit, sum_near, cnt_near, 0.5f);
}